// S4LinearModelV1_74569222193355
// MI455X (gfx1250) — compile-verified
//
#include <hip/hip_runtime.h>
#include <math.h>

#define BB 8
#define LL 16384
#define HH 64
#define N2 32
#define DEPTH 4
#define CH 1024
#define NC (LL / CH)          // 16 chunks
#define EPSV 1e-5f

typedef float v2f __attribute__((ext_vector_type(2)));
typedef float v8f __attribute__((ext_vector_type(8)));

// ---------------------------------------------------------------------------
// K0: s = 20*log10(|x|+eps); h = tanh(s*w_in + b_in)   -> [B,L,H]
// ---------------------------------------------------------------------------
__global__ void k_input(const float* __restrict__ x, const float* __restrict__ w_in,
                        const float* __restrict__ b_in, float* __restrict__ out) {
  int gid = blockIdx.x * blockDim.x + threadIdx.x;    // over B*L*H
  int h = gid & (HH - 1);
  int bl = gid >> 6;
  float xv = x[bl];
  float s = 20.0f * __log10f(fabsf(xv) + EPSV);
  out[gid] = tanhf(fmaf(s, w_in[h], b_in[h]));
}

// ---------------------------------------------------------------------------
// per-(h,n) discrete SSM pole: w = exp(dt*A), A = -exp(logAre) + i*Aim
// ---------------------------------------------------------------------------
__device__ __forceinline__ void ssm_pole(const float* log_dt, const float* log_A_real,
                                         const float* A_imag, int h, int n,
                                         float& dre, float& dim, float& wr, float& wi,
                                         float& Are, float& Aim) {
  float dt = __expf(log_dt[h]);
  Are = -__expf(log_A_real[h * N2 + n]);
  Aim = A_imag[h * N2 + n];
  dre = dt * Are;
  dim = dt * Aim;
  float er = __expf(dre);
  wr = er * __cosf(dim);
  wi = er * __sinf(dim);
}

// ---------------------------------------------------------------------------
// K1: phase-1 blocked scan — per-chunk final state with zero init.
// one wave per (b,h,chunk); lane = complex mode n
// ---------------------------------------------------------------------------
__global__ void k_scan1(const float* __restrict__ u, const float* __restrict__ log_dt,
                        const float* __restrict__ log_A_real,
                        const float* __restrict__ A_imag, float2* __restrict__ sfin) {
  int wid = (blockIdx.x * blockDim.x + threadIdx.x) >> 5;   // (b*H + h)*NC + c
  int lane = threadIdx.x & 31;
  int c = wid & (NC - 1);
  int h = (wid >> 4) & (HH - 1);
  int b = wid >> 10;
  float dre, dim, wr, wi, Are, Aim;
  ssm_pole(log_dt, log_A_real, A_imag, h, lane, dre, dim, wr, wi, Are, Aim);
  float sre = 0.f, sim = 0.f;
  const float* up = u + ((size_t)b * LL + (size_t)c * CH) * HH + h;
  for (int l0 = 0; l0 < CH; l0 += 32) {
    float uv = up[(size_t)(l0 + lane) * HH];
    if (l0 + 32 < CH) __builtin_prefetch(&up[(size_t)(l0 + 32 + lane) * HH], 0, 1);
#pragma unroll
    for (int t = 0; t < 32; ++t) {
      float ub = __shfl(uv, t, 32);
      float nr = fmaf(wr, sre, fmaf(-wi, sim, ub));
      float ni = fmaf(wi, sre, wr * sim);
      sre = nr; sim = ni;
    }
  }
  sfin[(size_t)wid * 32 + lane] = make_float2(sre, sim);
}

// ---------------------------------------------------------------------------
// K2: sequential combine over chunks: init[c] = w^CH * init[c-1] + sfin[c-1]
// one thread per (b,h,n)
// ---------------------------------------------------------------------------
__global__ void k_combine(const float* __restrict__ log_dt,
                          const float* __restrict__ log_A_real,
                          const float* __restrict__ A_imag,
                          const float2* __restrict__ sfin, float2* __restrict__ sinit) {
  int gid = blockIdx.x * blockDim.x + threadIdx.x;    // over B*H*N2
  int n = gid & 31;
  int h = (gid >> 5) & (HH - 1);
  int b = gid >> 11;
  float dre, dim, wr, wi, Are, Aim;
  ssm_pole(log_dt, log_A_real, A_imag, h, n, dre, dim, wr, wi, Are, Aim);
  // W = w^CH : magnitude exact, angle reduced in double to keep phase sane
  float Wmag = __expf((float)CH * dre);
  double ang = fmod((double)CH * (double)dim, 6.283185307179586);
  float Wr = Wmag * __cosf((float)ang);
  float Wi = Wmag * __sinf((float)ang);
  float ire = 0.f, iim = 0.f;
  size_t base = ((size_t)(b * HH + h) * NC) * 32 + n;
  for (int c = 0; c < NC; ++c) {
    sinit[base + (size_t)c * 32] = make_float2(ire, iim);
    float2 sf = sfin[base + (size_t)c * 32];
    float nr = fmaf(Wr, ire, fmaf(-Wi, iim, sf.x));
    float ni = fmaf(Wi, ire, fmaf(Wr, iim, sf.y));
    ire = nr; iim = ni;
  }
}

// ---------------------------------------------------------------------------
// K3: phase-2 scan with correct initial state; emits y = tanh(2*Re(Ceff.s)+D*u)
// butterfly shfl_xor reduction over the 32 modes per timestep
// ---------------------------------------------------------------------------
__global__ void k_scan2(const float* __restrict__ u, const float* __restrict__ log_dt,
                        const float* __restrict__ log_A_real,
                        const float* __restrict__ A_imag,
                        const float* __restrict__ C_re, const float* __restrict__ C_im,
                        const float* __restrict__ Dp,
                        const float2* __restrict__ sinit, float* __restrict__ out) {
  int wid = (blockIdx.x * blockDim.x + threadIdx.x) >> 5;
  int lane = threadIdx.x & 31;
  int c = wid & (NC - 1);
  int h = (wid >> 4) & (HH - 1);
  int b = wid >> 10;
  float dre, dim, wr, wi, Are, Aim;
  ssm_pole(log_dt, log_A_real, A_imag, h, lane, dre, dim, wr, wi, Are, Aim);
  // C_eff = (C_re + i C_im) * (w - 1) / A
  float Er = wr - 1.0f, Ei = wi;
  float inv = 1.0f / fmaf(Are, Are, Aim * Aim);
  float Fr = fmaf(Er, Are, Ei * Aim) * inv;
  float Fi = fmaf(Ei, Are, -Er * Aim) * inv;
  float Crr = C_re[h * N2 + lane], Cii = C_im[h * N2 + lane];
  float Gr = Crr * Fr - Cii * Fi;
  float Gi = Crr * Fi + Cii * Fr;
  float Dv = Dp[h];
  float2 s0 = sinit[(size_t)wid * 32 + lane];
  float sre = s0.x, sim = s0.y;
  const float* up = u + ((size_t)b * LL + (size_t)c * CH) * HH + h;
  float* op = out + ((size_t)b * LL + (size_t)c * CH) * HH + h;
  for (int l0 = 0; l0 < CH; l0 += 32) {
    float uv = up[(size_t)(l0 + lane) * HH];
    if (l0 + 32 < CH) __builtin_prefetch(&up[(size_t)(l0 + 32 + lane) * HH], 0, 1);
    float ymine = 0.f;
#pragma unroll
    for (int t = 0; t < 32; ++t) {
      float ub = __shfl(uv, t, 32);
      float nr = fmaf(wr, sre, fmaf(-wi, sim, ub));
      float ni = fmaf(wi, sre, wr * sim);
      sre = nr; sim = ni;
      float part = fmaf(Gr, sre, -Gi * sim);
      part += __shfl_xor(part, 1, 32);
      part += __shfl_xor(part, 2, 32);
      part += __shfl_xor(part, 4, 32);
      part += __shfl_xor(part, 8, 32);
      part += __shfl_xor(part, 16, 32);
      if (t == lane) ymine = part;     // lane l%32 keeps y[l]
    }
    float yl = fmaf(Dv, uv, 2.0f * ymine);
    op[(size_t)(l0 + lane) * HH] = tanhf(yl);
  }
}

// ---------------------------------------------------------------------------
// K4: mid linear  hout = tanh(hin @ W^T + b)   M=131072, K=N=64
// fp32 WMMA 16x16x4; block = 256 thr = 8 waves = 2(M-tiles) x 4(N-tiles)
// ---------------------------------------------------------------------------
#if __has_builtin(__builtin_amdgcn_wmma_f32_16x16x4_f32)
__global__ void k_mid(const float* __restrict__ hin, const float* __restrict__ w,
                      const float* __restrict__ bias, float* __restrict__ hout) {
  __shared__ float sA[32 * 66];   // pad to 66 -> conflict-free over 64 banks
  __shared__ float sW[64 * 66];
  int tid = threadIdx.x;
  int m0 = blockIdx.x * 32;
  for (int idx = tid; idx < 64 * 64; idx += 256) {
    sW[(idx >> 6) * 66 + (idx & 63)] = w[idx];
  }
  const float* hb = hin + (size_t)m0 * HH;
  for (int idx = tid; idx < 32 * 64; idx += 256) {
    sA[(idx >> 6) * 66 + (idx & 63)] = hb[idx];
  }
  __syncthreads();
  int wv = tid >> 5, lane = tid & 31;
  int tm = (wv >> 2) * 16;          // M-tile base within block
  int tn = (wv & 3) * 16;           // N-tile base
  int l15 = lane & 15;
  int koff = (lane >> 4) * 2;       // lanes 16-31 carry K=2,3 of each K-quad
  const float* arow = &sA[(tm + l15) * 66 + koff];
  const float* brow = &sW[(tn + l15) * 66 + koff];   // B[k][n] = W[n][k]
  v8f acc = {0.f, 0.f, 0.f, 0.f, 0.f, 0.f, 0.f, 0.f};
#pragma unroll
  for (int k0 = 0; k0 < 64; k0 += 4) {
    v2f a = *(const v2f*)(arow + k0);
    v2f bf = *(const v2f*)(brow + k0);
    acc = __builtin_amdgcn_wmma_f32_16x16x4_f32(false, a, false, bf,
                                                (short)0, acc, false, false);
  }
  // C/D layout: vgpr r, lanes 0-15 -> M=r, lanes 16-31 -> M=8+r; N = lane&15
  int ncol = tn + l15;
  int rbase = tm + ((lane >> 4) << 3);
  float bc = bias[ncol];
#pragma unroll
  for (int r = 0; r < 8; ++r) {
    hout[(size_t)(m0 + rbase + r) * HH + ncol] = tanhf(acc[r] + bc);
  }
}
#else
// scalar fallback (keeps this round compiling if the f32x4 builtin is absent)
__global__ void k_mid(const float* __restrict__ hin, const float* __restrict__ w,
                      const float* __restrict__ bias, float* __restrict__ hout) {
  int tid = threadIdx.x;
  int m0 = blockIdx.x * 32;
  for (int e = tid; e < 32 * 64; e += 256) {
    int row = m0 + (e >> 6), col = e & 63;
    const float* hr = hin + (size_t)row * HH;
    const float* wr = w + (size_t)col * HH;
    float acc = 0.f;
#pragma unroll
    for (int k = 0; k < HH; ++k) acc = fmaf(hr[k], wr[k], acc);
    hout[(size_t)row * HH + col] = tanhf(acc + bias[col]);
  }
}
#endif

// ---------------------------------------------------------------------------
// K5: g = h.w_out + b_out ; out = x * 10^(g/20)
// ---------------------------------------------------------------------------
__global__ void k_out(const float* __restrict__ hin, const float* __restrict__ x,
                      const float* __restrict__ w_out, const float* __restrict__ b_out,
                      float* __restrict__ out) {
  int gid = blockIdx.x * blockDim.x + threadIdx.x;     // over B*L
  const float* hr = hin + (size_t)gid * HH;
  float acc = 0.f;
#pragma unroll
  for (int k = 0; k < HH; k += 4) {
    float4 hv = *(const float4*)(hr + k);
    float4 wv = *(const float4*)(w_out + k);
    acc = fmaf(hv.x, wv.x, acc);
    acc = fmaf(hv.y, wv.y, acc);
    acc = fmaf(hv.z, wv.z, acc);
    acc = fmaf(hv.w, wv.w, acc);
  }
  float g = acc + b_out[0];
  out[gid] = x[gid] * __expf(g * 0.11512925464970229f);   // ln(10)/20
}

// ---------------------------------------------------------------------------
extern "C" void kernel_launch(void* const* d_in, const int* in_sizes, int n_in,
                              void* d_out, int out_size, void* d_ws, size_t ws_size,
                              hipStream_t stream) {
  const float* x      = (const float*)d_in[0];
  const float* w_in   = (const float*)d_in[1];
  const float* b_in   = (const float*)d_in[2];
  const float* w_mid  = (const float*)d_in[3];
  const float* b_mid  = (const float*)d_in[4];
  const float* w_out  = (const float*)d_in[5];
  const float* b_out  = (const float*)d_in[6];
  const float* log_dt = (const float*)d_in[7];
  const float* log_A  = (const float*)d_in[8];
  const float* A_imag = (const float*)d_in[9];
  const float* C_re   = (const float*)d_in[10];
  const float* C_im   = (const float*)d_in[11];
  const float* Dp     = (const float*)d_in[12];
  float* out = (float*)d_out;

  const size_t nBLH = (size_t)BB * LL * HH;               // 8.39M floats (32 MB)
  float*  buf0  = (float*)d_ws;
  float*  buf1  = buf0 + nBLH;
  float2* sfin  = (float2*)(buf1 + nBLH);                 // [B,H,NC,N2] complex
  float2* sinit = sfin + (size_t)BB * HH * NC * N2;       // [B,H,NC,N2] complex

  k_input<<<(int)(nBLH / 256), 256, 0, stream>>>(x, w_in, b_in, buf0);

  auto dssm = [&](const float* uin, float* uout, int j) {
    const float* ld = log_dt + j * HH;
    const float* la = log_A + (size_t)j * HH * N2;
    const float* ai = A_imag + (size_t)j * HH * N2;
    k_scan1<<<(BB * HH * NC * 32) / 256, 256, 0, stream>>>(uin, ld, la, ai, sfin);
    k_combine<<<(BB * HH * N2) / 256, 256, 0, stream>>>(ld, la, ai, sfin, sinit);
    k_scan2<<<(BB * HH * NC * 32) / 256, 256, 0, stream>>>(
        uin, ld, la, ai, C_re + (size_t)j * HH * N2, C_im + (size_t)j * HH * N2,
        Dp + j * HH, sinit, uout);
  };

  dssm(buf0, buf1, 0);
  for (int i = 0; i < DEPTH; ++i) {
    k_mid<<<(BB * LL) / 32, 256, 0, stream>>>(buf1, w_mid + (size_t)i * HH * HH,
                                              b_mid + i * HH, buf0);
    dssm(buf0, buf1, i + 1);
  }
  k_out<<<(BB * LL) / 256, 256, 0, stream>>>(buf1, x, w_out, b_out, out);
}